// HTR_SparseAtt_77077483094578
// MI455X (gfx1250) — compile-verified
//
#include <hip/hip_runtime.h>
#include <hip/hip_bf16.h>
#include <math.h>

// ---------------------------------------------------------------------------
// Problem constants (reference: C=64, H=W=512, fp32 everywhere)
// ---------------------------------------------------------------------------
#define CC 64
#define HH 512
#define WW 512
#define HWSZ (HH * WW)                       // 262144
#define PLANE ((size_t)CC * HWSZ)            // 16,777,216 floats (64 MB)
#define EPSV 1e-5f

typedef float v2f __attribute__((ext_vector_type(2)));
typedef float v8f __attribute__((ext_vector_type(8)));

// ---------------------------------------------------------------------------
// 1-D min/max pool (square max-pool is separable; -inf/+inf identity via
// skipping out-of-range taps, matching jax reduce_window's -inf padding).
// ---------------------------------------------------------------------------
template <bool HORIZ, bool ISMIN>
__global__ void pool1d_k(const float* __restrict__ src, float* __restrict__ dst, int K) {
    size_t idx = (size_t)blockIdx.x * blockDim.x + threadIdx.x;
    if (idx >= PLANE) return;
    int w = (int)(idx % WW);
    int h = (int)((idx / WW) % HH);
    int p = K >> 1;
    float acc = ISMIN ? INFINITY : -INFINITY;
    if (HORIZ) {
        int lo = w - p; if (lo < 0) lo = 0;
        int hi = w + p; if (hi > WW - 1) hi = WW - 1;
        const float* row = src + (idx - (size_t)w);
        for (int x = lo; x <= hi; ++x) {
            float v = row[x];
            acc = ISMIN ? fminf(acc, v) : fmaxf(acc, v);
        }
    } else {
        int lo = h - p; if (lo < 0) lo = 0;
        int hi = h + p; if (hi > HH - 1) hi = HH - 1;
        size_t base = idx - (size_t)h * WW;
        for (int y = lo; y <= hi; ++y) {
            float v = src[base + (size_t)y * WW];
            acc = ISMIN ? fminf(acc, v) : fmaxf(acc, v);
        }
    }
    dst[idx] = acc;
}

// ---------------------------------------------------------------------------
// Depthwise KxK conv, zero padding ('same').
// ---------------------------------------------------------------------------
template <int K>
__global__ void dwconv_k(const float* __restrict__ src, const float* __restrict__ wgt,
                         const float* __restrict__ bias, float* __restrict__ dst) {
    size_t idx = (size_t)blockIdx.x * blockDim.x + threadIdx.x;
    if (idx >= PLANE) return;
    int w = (int)(idx % WW);
    int h = (int)((idx / WW) % HH);
    int c = (int)(idx / HWSZ);
    const float* wk = wgt + (size_t)c * K * K;
    const float* pl = src + (size_t)c * HWSZ;
    constexpr int P = K / 2;
    float acc = bias[c];
#pragma unroll
    for (int dy = 0; dy < K; ++dy) {
        int y = h + dy - P;
        if (y < 0 || y >= HH) continue;
#pragma unroll
        for (int dx = 0; dx < K; ++dx) {
            int x = w + dx - P;
            if (x < 0 || x >= WW) continue;
            acc += wk[dy * K + dx] * pl[(size_t)y * WW + x];
        }
    }
    dst[idx] = acc;
}

// ---------------------------------------------------------------------------
// Per-channel plane stats: mu and rsqrt(var+eps). grid.x = #channels.
// ---------------------------------------------------------------------------
__global__ void stats_k(const float* __restrict__ src, float* __restrict__ mu,
                        float* __restrict__ rs) {
    int c = blockIdx.x;
    const float* p = src + (size_t)c * HWSZ;
    float s = 0.f, s2 = 0.f;
    for (int i = threadIdx.x; i < HWSZ; i += blockDim.x) {
        float v = p[i];
        s += v;
        s2 += v * v;
    }
    __shared__ float sh[256], sh2[256];
    sh[threadIdx.x] = s;
    sh2[threadIdx.x] = s2;
    __syncthreads();
    for (int off = 128; off > 0; off >>= 1) {
        if (threadIdx.x < off) {
            sh[threadIdx.x] += sh[threadIdx.x + off];
            sh2[threadIdx.x] += sh2[threadIdx.x + off];
        }
        __syncthreads();
    }
    if (threadIdx.x == 0) {
        float m = sh[0] * (1.0f / HWSZ);
        float var = sh2[0] * (1.0f / HWSZ) - m * m;
        mu[c] = m;
        rs[c] = rsqrtf(fmaxf(var, 0.f) + EPSV);
    }
}

// ---------------------------------------------------------------------------
// diff = sign * (x - relu((pre - mu)*rs))   (sign=+1: WTHAM, -1: BTHAM)
// ---------------------------------------------------------------------------
__global__ void diff_k(const float* __restrict__ x, const float* __restrict__ pre,
                       const float* __restrict__ mu, const float* __restrict__ rs,
                       float sign, float* __restrict__ diff) {
    size_t idx = (size_t)blockIdx.x * blockDim.x + threadIdx.x;
    if (idx >= PLANE) return;
    int c = (int)(idx / HWSZ);
    float v = fmaxf((pre[idx] - mu[c]) * rs[c], 0.f);
    diff[idx] = sign * (x[idx] - v);
}

// ---------------------------------------------------------------------------
// combined (+)= dwconv3x3(InstanceNorm(diff)) + bias, with IN folded through
// the depthwise conv as per-channel alpha/beta. Zero-padding handled exactly
// by tracking the valid-tap weight sum (beta multiplies only in-image taps).
// ---------------------------------------------------------------------------
__global__ void conv3_acc_k(const float* __restrict__ diff, const float* __restrict__ w3,
                            const float* __restrict__ b3, const float* __restrict__ mu,
                            const float* __restrict__ rs, float* __restrict__ combined,
                            int accum) {
    size_t idx = (size_t)blockIdx.x * blockDim.x + threadIdx.x;
    if (idx >= PLANE) return;
    int w = (int)(idx % WW);
    int h = (int)((idx / WW) % HH);
    int c = (int)(idx / HWSZ);
    float alpha = rs[c];
    float beta = -mu[c] * alpha;
    const float* wk = w3 + (size_t)c * 9;
    const float* pl = diff + (size_t)c * HWSZ;
    float av = 0.f, aw = 0.f;
#pragma unroll
    for (int dy = 0; dy < 3; ++dy) {
        int y = h + dy - 1;
        if (y < 0 || y >= HH) continue;
#pragma unroll
        for (int dx = 0; dx < 3; ++dx) {
            int x = w + dx - 1;
            if (x < 0 || x >= WW) continue;
            float wv = wk[dy * 3 + dx];
            av += wv * pl[(size_t)y * WW + x];
            aw += wv;
        }
    }
    float out = alpha * av + beta * aw + b3[c];
    combined[idx] = accum ? (combined[idx] + out) : out;
}

// ---------------------------------------------------------------------------
// Conv3d(1->3, 3x3x3, pad 1) as implicit GEMM on V_WMMA_F32_16X16X4_F32.
//   D[m,n] = sum_k A[m,k] * B[k,n]
//   m = output channel (rows 0..2 live, 3..15 zero), n = 16 output pixels,
//   k = 27 taps processed in 7 chunks of 4 (tap 27 zero-padded).
// A layout (16x4 f32): lanes 0-15 -> K=k0,k0+1 in VGPR0/1; lanes 16-31 -> K=k0+2,k0+3.
// B layout mirrors A; D rows 0..2 land in c[0..2] of lanes 0..15.
//
// Padding is BRANCHLESS: always load from a clamped in-bounds address, then
// v_cndmask-select 0 for out-of-range taps. No EXEC manipulation inside the
// loop -> loads can clause, no per-tap s_wait_xcnt, EXEC provably all-ones
// at every WMMA.
// ---------------------------------------------------------------------------
__device__ __forceinline__ int clampi(int v, int lo, int hi) {
    v = v < lo ? lo : v;
    return v > hi ? hi : v;
}

__device__ __forceinline__ float fetch_tap(const float* __restrict__ src, int d, int h,
                                           int w, int kk) {
    int kkc = kk < 27 ? kk : 0;                 // clamp tap index (tap 27 is padding)
    int kz = kkc / 9;
    int kr = kkc - kz * 9;
    int ky = kr / 3;
    int kx = kr - ky * 3;
    int z = d + kz - 1;
    int y = h + ky - 1;
    int x = w + kx - 1;
    bool inb = (kk < 27) & ((unsigned)z < (unsigned)CC) & ((unsigned)y < (unsigned)HH) &
               ((unsigned)x < (unsigned)WW);
    int zc = clampi(z, 0, CC - 1);
    int yc = clampi(y, 0, HH - 1);
    int xc = clampi(x, 0, WW - 1);
    float v = src[((size_t)zc * HH + yc) * WW + xc];  // always in-bounds
    return inb ? v : 0.f;                             // cndmask, no branch
}

__global__ void conv3d1_wmma_k(const float* __restrict__ src,   // [64,512,512]
                               const float* __restrict__ wt,    // [3,1,3,3,3] flat
                               const float* __restrict__ bt,    // [3]
                               float* __restrict__ out) {       // [3,64,512,512]
    int wave = (int)((((size_t)blockIdx.x * blockDim.x) + threadIdx.x) >> 5);
    int lane = (int)(threadIdx.x & 31);
    const int TPR = WW / 16;                  // 32 tiles per row
    int tw = wave % TPR;
    int th = (wave / TPR) % HH;
    int td = wave / (TPR * HH);
    int w0 = tw * 16;
    int n = lane & 15;
    int half = lane >> 4;
    int wpix = w0 + n;

    v8f acc = {};
#pragma unroll
    for (int chunk = 0; chunk < 7; ++chunk) {
        int k0 = chunk * 4 + 2 * half;
        // A: weights (row m = lane&15; only m<3 nonzero) -- branchless selects
        bool mlive = n < 3;
        int wrow = mlive ? n : 0;
        float a0 = (mlive & (k0 < 27)) ? wt[wrow * 27 + clampi(k0, 0, 26)] : 0.f;
        float a1 = (mlive & (k0 + 1 < 27)) ? wt[wrow * 27 + clampi(k0 + 1, 0, 26)] : 0.f;
        v2f a; a.x = a0; a.y = a1;
        // B: input taps for this pixel (branchless clamped loads)
        v2f b;
        b.x = fetch_tap(src, td, th, wpix, k0);
        b.y = fetch_tap(src, td, th, wpix, k0 + 1);
        acc = __builtin_amdgcn_wmma_f32_16x16x4_f32(false, a, false, b, (short)0, acc,
                                                    false, false);
    }
    if (half == 0) {
        size_t base = ((size_t)td * HH + th) * WW + wpix;
        size_t plane3 = (size_t)CC * HH * WW;
        out[base] = acc[0] + bt[0];
        out[base + plane3] = acc[1] + bt[1];
        out[base + 2 * plane3] = acc[2] + bt[2];
    }
}

// ---------------------------------------------------------------------------
// In-place relu(InstanceNorm) over 192 (oc,d) planes of t3.
// ---------------------------------------------------------------------------
__global__ void norm_relu_k(float* __restrict__ t, const float* __restrict__ mu,
                            const float* __restrict__ rs) {
    size_t idx = (size_t)blockIdx.x * blockDim.x + threadIdx.x;
    if (idx >= 3 * PLANE) return;
    int ch = (int)(idx / HWSZ);
    t[idx] = fmaxf((t[idx] - mu[ch]) * rs[ch], 0.f);
}

// ---------------------------------------------------------------------------
// Conv3d(3->1, 3x3x3, pad 1): 81-tap VALU (single output channel; not
// matrix-shaped enough to benefit from WMMA).
// ---------------------------------------------------------------------------
__global__ void conv3d2_k(const float* __restrict__ tn, const float* __restrict__ wt,
                          const float* __restrict__ b2, float* __restrict__ out) {
    size_t idx = (size_t)blockIdx.x * blockDim.x + threadIdx.x;
    if (idx >= PLANE) return;
    int w = (int)(idx % WW);
    int h = (int)((idx / WW) % HH);
    int d = (int)(idx / HWSZ);
    size_t plane3 = PLANE;
    float acc = b2[0];
#pragma unroll
    for (int ic = 0; ic < 3; ++ic) {
        const float* base = tn + (size_t)ic * plane3;
        const float* wk = wt + ic * 27;
#pragma unroll
        for (int dz = 0; dz < 3; ++dz) {
            int z = d + dz - 1;
            if (z < 0 || z >= CC) continue;
#pragma unroll
            for (int dy = 0; dy < 3; ++dy) {
                int y = h + dy - 1;
                if (y < 0 || y >= HH) continue;
#pragma unroll
                for (int dx = 0; dx < 3; ++dx) {
                    int x = w + dx - 1;
                    if (x < 0 || x >= WW) continue;
                    acc += wk[dz * 9 + dy * 3 + dx] * base[((size_t)z * HH + y) * WW + x];
                }
            }
        }
    }
    out[idx] = acc;
}

// ---------------------------------------------------------------------------
// y = relu(x * sigmoid(t))
// ---------------------------------------------------------------------------
__global__ void gate_k(const float* __restrict__ x, const float* __restrict__ t,
                       float* __restrict__ y) {
    size_t idx = (size_t)blockIdx.x * blockDim.x + threadIdx.x;
    if (idx >= PLANE) return;
    float s = 1.f / (1.f + expf(-t[idx]));
    y[idx] = fmaxf(x[idx] * s, 0.f);
}

// ---------------------------------------------------------------------------
// Host-side stage driver (templated on morphology kernel size K)
// ---------------------------------------------------------------------------
static constexpr int EW_BLOCKS = (int)(PLANE / 256);       // 65536
static constexpr int EW3_BLOCKS = (int)(3 * PLANE / 256);  // 196608

template <int K>
static void run_stage(const float* x, const float* w_w, const float* b_w,
                      const float* w_b, const float* b_b, const float* w_wn,
                      const float* b_wn, float* combined, float* t0, float* t1,
                      float* mu, float* rs, int accumFirst, hipStream_t s) {
    // ---- WTHAM: closing = erode(dilate(x)) -> dwconv -> IN -> ReLU ----
    pool1d_k<true, false><<<EW_BLOCKS, 256, 0, s>>>(x, t0, K);
    pool1d_k<false, false><<<EW_BLOCKS, 256, 0, s>>>(t0, t1, K);
    pool1d_k<true, true><<<EW_BLOCKS, 256, 0, s>>>(t1, t0, K);
    pool1d_k<false, true><<<EW_BLOCKS, 256, 0, s>>>(t0, t1, K);     // t1 = closing
    dwconv_k<K><<<EW_BLOCKS, 256, 0, s>>>(t1, w_w, b_w, t0);        // t0 = pre
    stats_k<<<CC, 256, 0, s>>>(t0, mu, rs);
    diff_k<<<EW_BLOCKS, 256, 0, s>>>(x, t0, mu, rs, 1.f, t1);       // t1 = x - wth
    stats_k<<<CC, 256, 0, s>>>(t1, mu, rs);
    conv3_acc_k<<<EW_BLOCKS, 256, 0, s>>>(t1, w_wn, b_wn, mu, rs, combined, accumFirst);
    // ---- BTHAM: opening = dilate(erode(x)) -> dwconv -> IN -> ReLU ----
    pool1d_k<true, true><<<EW_BLOCKS, 256, 0, s>>>(x, t0, K);
    pool1d_k<false, true><<<EW_BLOCKS, 256, 0, s>>>(t0, t1, K);
    pool1d_k<true, false><<<EW_BLOCKS, 256, 0, s>>>(t1, t0, K);
    pool1d_k<false, false><<<EW_BLOCKS, 256, 0, s>>>(t0, t1, K);    // t1 = opening
    dwconv_k<K><<<EW_BLOCKS, 256, 0, s>>>(t1, w_b, b_b, t0);        // t0 = pre
    stats_k<<<CC, 256, 0, s>>>(t0, mu, rs);
    diff_k<<<EW_BLOCKS, 256, 0, s>>>(x, t0, mu, rs, -1.f, t1);      // t1 = bth - x
    stats_k<<<CC, 256, 0, s>>>(t1, mu, rs);
    conv3_acc_k<<<EW_BLOCKS, 256, 0, s>>>(t1, w_wn, b_wn, mu, rs, combined, 1);
}

extern "C" void kernel_launch(void* const* d_in, const int* in_sizes, int n_in,
                              void* d_out, int out_size, void* d_ws, size_t ws_size,
                              hipStream_t stream) {
    (void)in_sizes; (void)n_in; (void)out_size; (void)ws_size;
    // Inputs, in setup_inputs() order:
    const float* x     = (const float*)d_in[0];
    const float* w_wt5 = (const float*)d_in[1];
    const float* b_wt5 = (const float*)d_in[2];
    const float* w_wt9 = (const float*)d_in[3];
    const float* b_wt9 = (const float*)d_in[4];
    const float* w_bt5 = (const float*)d_in[5];
    const float* b_bt5 = (const float*)d_in[6];
    const float* w_bt9 = (const float*)d_in[7];
    const float* b_bt9 = (const float*)d_in[8];
    const float* w_wn  = (const float*)d_in[9];
    const float* b_wn  = (const float*)d_in[10];
    const float* w_t1  = (const float*)d_in[11];
    const float* b_t1  = (const float*)d_in[12];
    const float* w_t2  = (const float*)d_in[13];
    const float* b_t2  = (const float*)d_in[14];

    // Workspace layout: combined | t0 | t1 | t3(3 planes) | stats(384 floats)
    float* ws = (float*)d_ws;
    float* combined = ws;
    float* t0 = ws + PLANE;
    float* t1 = ws + 2 * PLANE;
    float* t3 = ws + 3 * PLANE;
    float* mu = ws + 6 * PLANE;
    float* rs = mu + 192;
    float* t2 = t0;  // free after combined is finished

    run_stage<5>(x, w_wt5, b_wt5, w_bt5, b_bt5, w_wn, b_wn, combined, t0, t1, mu, rs,
                 /*accumFirst=*/0, stream);
    run_stage<9>(x, w_wt9, b_wt9, w_bt9, b_bt9, w_wn, b_wn, combined, t0, t1, mu, rs,
                 /*accumFirst=*/1, stream);

    // Temporal cross: Conv3d(1->3) via WMMA f32 16x16x4 implicit GEMM.
    // waves = 64 * 512 * 32 = 1,048,576; 8 waves/block -> 131072 blocks (exact).
    conv3d1_wmma_k<<<131072, 256, 0, stream>>>(combined, w_t1, b_t1, t3);
    stats_k<<<192, 256, 0, stream>>>(t3, mu, rs);
    norm_relu_k<<<EW3_BLOCKS, 256, 0, stream>>>(t3, mu, rs);
    conv3d2_k<<<EW_BLOCKS, 256, 0, stream>>>(t3, w_t2, b_t2, t2);
    gate_k<<<EW_BLOCKS, 256, 0, stream>>>(x, t2, (float*)d_out);
}